// GraphAttentionNetwork_72249939853604
// MI455X (gfx1250) — compile-verified
//
#include <hip/hip_runtime.h>
#include <math.h>

// GAT, 2 layers, for MI455X (gfx1250, wave32).
// GEMMs use V_WMMA_F32_16X16X4_F32 (fp32 matches the fp32 reference; the
// workload is memory-bound on the ~1GB edge gather/scatter, so low-precision
// WMMA buys nothing). Segment softmax via ordered-uint atomicMax + f32
// atomicAdd. GEMM dims are template constants so all strides fold into
// immediate offsets; each wave computes a 16x64 C macro-tile with 4
// independent WMMA accumulator chains.

#define N_NODES 10000
#define F_IN    64
#define HID     128
#define HEADS   4
#define OUT_C   256
#define NEG_SLOPE 0.2f
#define GAT_EPS 1e-16f

typedef __attribute__((ext_vector_type(2))) float v2f;
typedef __attribute__((ext_vector_type(8))) float v8f;

// ---------------- WMMA fp32 GEMM: C[M,Nc] = A[M,K] * B[K,Nc] ----------------
// One wave computes a 16x64 tile of C (4 16x16 WMMA tiles along N), stepping
// K by 4.  Layouts per ISA 7.12.2:
//  A 16x4 : lanes 0-15 -> M=lane, VGPR{0,1}=K{k0,k0+1};
//           lanes 16-31 -> M=lane-16, VGPR{0,1}=K{k0+2,k0+3}.
//  B 4x16 : VGPR j -> row K=k0+j (lanes 0-15) / K=k0+j+2 (lanes 16-31),
//           N = lane&15  (B rows striped across lanes like C/D).
//  C 16x16: VGPR r -> M = tm + r + 8*(lane>=16), N = tn + (lane&15).
template <int K, int Nc>
__global__ __launch_bounds__(32) void wmma_gemm_f32(
    const float* __restrict__ A, const float* __restrict__ B,
    float* __restrict__ C) {
  constexpr int WN = 64;               // C columns per wave
  constexpr int waves_n = Nc / WN;
  const int tile = blockIdx.x;
  const int tm = (tile / waves_n) << 4;
  const int tn = (tile % waves_n) * WN;
  const int lane = threadIdx.x & 31;
  const int half = lane >> 4;          // 0: lanes 0-15, 1: lanes 16-31
  const int l    = lane & 15;

  v8f acc0 = {}, acc1 = {}, acc2 = {}, acc3 = {};
  const float* Arow  = A + (size_t)(tm + l) * K + 2 * half;  // +k0 -> K pair
  const float* Bbase = B + tn + l;

#pragma unroll 4
  for (int k0 = 0; k0 < K; k0 += 4) {
    v2f a;
    a.x = Arow[k0];
    a.y = Arow[k0 + 1];
    const float* Br0 = Bbase + (size_t)(k0 + 2 * half) * Nc;  // K row (this half)
    const float* Br1 = Br0 + Nc;                              // next K row
    v2f b0 = {Br0[0],  Br1[0]};
    v2f b1 = {Br0[16], Br1[16]};
    v2f b2 = {Br0[32], Br1[32]};
    v2f b3 = {Br0[48], Br1[48]};
    acc0 = __builtin_amdgcn_wmma_f32_16x16x4_f32(false, a, false, b0, (short)0, acc0, false, false);
    acc1 = __builtin_amdgcn_wmma_f32_16x16x4_f32(false, a, false, b1, (short)0, acc1, false, false);
    acc2 = __builtin_amdgcn_wmma_f32_16x16x4_f32(false, a, false, b2, (short)0, acc2, false, false);
    acc3 = __builtin_amdgcn_wmma_f32_16x16x4_f32(false, a, false, b3, (short)0, acc3, false, false);
  }

  float* Crow = C + (size_t)(tm + 8 * half) * Nc + tn + l;
#pragma unroll
  for (int r = 0; r < 8; ++r) {
    Crow[(size_t)r * Nc + 0]  = acc0[r];
    Crow[(size_t)r * Nc + 16] = acc1[r];
    Crow[(size_t)r * Nc + 32] = acc2[r];
    Crow[(size_t)r * Nc + 48] = acc3[r];
  }
}

// ---------------- attention projections: a_src/a_dst = <h[n,h,:], att[h,:]> --
__global__ void att_dots(const float* __restrict__ h,
                         const float* __restrict__ att_src,
                         const float* __restrict__ att_dst,
                         float* __restrict__ asrc, float* __restrict__ adst,
                         int heads, int ch) {
  int idx = blockIdx.x * blockDim.x + threadIdx.x;
  if (idx >= N_NODES * heads) return;
  int n = idx / heads, hh = idx % heads;
  const float* hv = h + (size_t)n * heads * ch + (size_t)hh * ch;
  const float* as = att_src + (size_t)hh * ch;
  const float* ad = att_dst + (size_t)hh * ch;
  float s = 0.f, d = 0.f;
  for (int c = 0; c < ch; ++c) { float v = hv[c]; s += v * as[c]; d += v * ad[c]; }
  asrc[idx] = s;
  adst[idx] = d;
}

// ---------------- ordered-uint mapping for float atomic max ----------------
__device__ __forceinline__ unsigned f2ord(float f) {
  unsigned u = __float_as_uint(f);
  return (u & 0x80000000u) ? ~u : (u | 0x80000000u);
}
__device__ __forceinline__ float ord2f(unsigned u) {
  return (u & 0x80000000u) ? __uint_as_float(u & 0x7fffffffu)
                           : __uint_as_float(~u);
}

// ---------------- pass 1: e = leakyrelu(a_src[s]+a_dst[d]); segment max -----
__global__ void edge_logits(const int* __restrict__ src, const int* __restrict__ dst,
                            int E0, int ET,
                            const float* __restrict__ asrc, const float* __restrict__ adst,
                            float* __restrict__ ebuf, unsigned* __restrict__ mmax,
                            int heads) {
  int idx = blockIdx.x * blockDim.x + threadIdx.x;
  if (idx >= ET * heads) return;
  int e = idx / heads, hh = idx % heads;
  int s = (e < E0) ? src[e] : (e - E0);   // self-loops appended after edges
  int d = (e < E0) ? dst[e] : (e - E0);
  float v = asrc[s * heads + hh] + adst[d * heads + hh];
  v = (v > 0.f) ? v : NEG_SLOPE * v;
  ebuf[idx] = v;
  atomicMax(&mmax[d * heads + hh], f2ord(v));
}

// ---------------- pass 2: p = exp(e - m[dst]); segment sum ------------------
__global__ void edge_exp(const int* __restrict__ src, const int* __restrict__ dst,
                         int E0, int ET,
                         float* __restrict__ ebuf, const unsigned* __restrict__ mmax,
                         float* __restrict__ denom, int heads) {
  int idx = blockIdx.x * blockDim.x + threadIdx.x;
  if (idx >= ET * heads) return;
  int e = idx / heads, hh = idx % heads;
  int d = (e < E0) ? dst[e] : (e - E0);
  float p = __expf(ebuf[idx] - ord2f(mmax[d * heads + hh]));
  ebuf[idx] = p;   // overwrite logits with unnormalized weights
  atomicAdd(&denom[d * heads + hh], p);
}

// ---------------- pass 3: out[d,:] += h[s,:] * alpha ------------------------
__global__ void edge_scatter(const int* __restrict__ src, const int* __restrict__ dst,
                             int E0, int ET,
                             const float* __restrict__ h, const float* __restrict__ p,
                             const float* __restrict__ denom, float* __restrict__ out,
                             int heads, int ch) {
  const int F = heads * ch;
  int idx = blockIdx.x * blockDim.x + threadIdx.x;   // ET*F <= 87.04M < 2^31
  if (idx >= ET * F) return;
  int e = idx / F, c = idx % F, hh = c / ch;
  int s = (e < E0) ? src[e] : (e - E0);
  int d = (e < E0) ? dst[e] : (e - E0);
  float alpha = p[e * heads + hh] / (denom[d * heads + hh] + GAT_EPS);
  atomicAdd(&out[(size_t)d * F + c], h[(size_t)s * F + c] * alpha);
}

// ---------------- bias + ELU (in place) -------------------------------------
__global__ void elu_bias(float* __restrict__ x, const float* __restrict__ bias,
                         int total, int F) {
  int idx = blockIdx.x * blockDim.x + threadIdx.x;
  if (idx >= total) return;
  float v = x[idx] + bias[idx % F];
  x[idx] = (v > 0.f) ? v : (__expf(v) - 1.f);
}

// ---------------- zero fill -------------------------------------------------
__global__ void zero_f(float* __restrict__ p, int n) {
  int idx = blockIdx.x * blockDim.x + threadIdx.x;
  if (idx < n) p[idx] = 0.f;
}

// ---------------- column mean + bias + broadcast to [rows, OUT_C] -----------
__global__ void mean_broadcast(const float* __restrict__ out2,
                               const float* __restrict__ b2,
                               float* __restrict__ y, int rows) {
  __shared__ float red[256];
  const int c = blockIdx.x;   // one block per output channel
  float s = 0.f;
  for (int n = threadIdx.x; n < N_NODES; n += blockDim.x)
    s += out2[(size_t)n * OUT_C + c];
  red[threadIdx.x] = s;
  __syncthreads();
  for (int off = 128; off > 0; off >>= 1) {
    if ((int)threadIdx.x < off) red[threadIdx.x] += red[threadIdx.x + off];
    __syncthreads();
  }
  if (threadIdx.x == 0) {
    float g = red[0] / (float)N_NODES + b2[c];
    for (int b = 0; b < rows; ++b) y[b * OUT_C + c] = g;
  }
}

// ---------------------------------------------------------------------------
extern "C" void kernel_launch(void* const* d_in, const int* in_sizes, int n_in,
                              void* d_out, int out_size, void* d_ws, size_t ws_size,
                              hipStream_t stream) {
  const int   E0       = in_sizes[0] / 2;
  const int*  src      = (const int*)d_in[0];
  const int*  dst      = src + E0;
  const float* node_emb = (const float*)d_in[2];
  const float* W1       = (const float*)d_in[3];
  const float* att_src1 = (const float*)d_in[4];
  const float* att_dst1 = (const float*)d_in[5];
  const float* b1       = (const float*)d_in[6];
  const float* W2       = (const float*)d_in[7];
  const float* att_src2 = (const float*)d_in[8];
  const float* att_dst2 = (const float*)d_in[9];
  const float* b2       = (const float*)d_in[10];
  float* out = (float*)d_out;

  const int ET   = E0 + N_NODES;       // with self-loops
  const int F1   = HEADS * HID;        // 512
  const int rows = out_size / OUT_C;   // batch_size

  // ---- workspace layout (floats) ----
  float* ws = (float*)d_ws;
  size_t off = 0;
  float* h1     = ws + off; off += (size_t)N_NODES * F1;   // 5.12M
  float* x1     = ws + off; off += (size_t)N_NODES * F1;   // 5.12M (aggregate -> ELU)
  float* asrc1  = ws + off; off += (size_t)N_NODES * HEADS;
  float* adst1  = ws + off; off += (size_t)N_NODES * HEADS;
  float* m1     = ws + off; off += (size_t)N_NODES * HEADS;   // ordered-uint
  float* denom1 = ws + off; off += (size_t)N_NODES * HEADS;
  float* p1     = ws + off; off += (size_t)ET * HEADS;
  float* asrc2  = ws + off; off += (size_t)N_NODES;
  float* adst2  = ws + off; off += (size_t)N_NODES;
  float* m2     = ws + off; off += (size_t)N_NODES;           // ordered-uint
  float* denom2 = ws + off; off += (size_t)N_NODES;
  float* p2     = ws + off; off += (size_t)ET;
  // h1 is dead after the layer-1 scatter -> alias layer-2 buffers over it.
  float* h2   = h1;                              // N*256
  float* out2 = h1 + (size_t)N_NODES * OUT_C;    // N*256

  const int TB = 256;
  auto blks = [](long long n, int b) { return (unsigned)((n + b - 1) / b); };

  // ================= layer 1 =================
  wmma_gemm_f32<F_IN, 512><<<(N_NODES / 16) * (512 / 64), 32, 0, stream>>>(
      node_emb, W1, h1);
  att_dots<<<blks((long long)N_NODES * HEADS, TB), TB, 0, stream>>>(
      h1, att_src1, att_dst1, asrc1, adst1, HEADS, HID);
  zero_f<<<blks(N_NODES * HEADS, TB), TB, 0, stream>>>(m1, N_NODES * HEADS);
  zero_f<<<blks(N_NODES * HEADS, TB), TB, 0, stream>>>(denom1, N_NODES * HEADS);
  zero_f<<<blks((long long)N_NODES * F1, TB), TB, 0, stream>>>(x1, N_NODES * F1);
  edge_logits<<<blks((long long)ET * HEADS, TB), TB, 0, stream>>>(
      src, dst, E0, ET, asrc1, adst1, p1, (unsigned*)m1, HEADS);
  edge_exp<<<blks((long long)ET * HEADS, TB), TB, 0, stream>>>(
      src, dst, E0, ET, p1, (const unsigned*)m1, denom1, HEADS);
  edge_scatter<<<blks((long long)ET * F1, TB), TB, 0, stream>>>(
      src, dst, E0, ET, h1, p1, denom1, x1, HEADS, HID);
  elu_bias<<<blks((long long)N_NODES * F1, TB), TB, 0, stream>>>(
      x1, b1, N_NODES * F1, F1);

  // ================= layer 2 =================
  wmma_gemm_f32<512, OUT_C><<<(N_NODES / 16) * (OUT_C / 64), 32, 0, stream>>>(
      x1, W2, h2);
  att_dots<<<blks(N_NODES, TB), TB, 0, stream>>>(
      h2, att_src2, att_dst2, asrc2, adst2, 1, OUT_C);
  zero_f<<<blks(N_NODES, TB), TB, 0, stream>>>(m2, N_NODES);
  zero_f<<<blks(N_NODES, TB), TB, 0, stream>>>(denom2, N_NODES);
  zero_f<<<blks((long long)N_NODES * OUT_C, TB), TB, 0, stream>>>(out2, N_NODES * OUT_C);
  edge_logits<<<blks(ET, TB), TB, 0, stream>>>(
      src, dst, E0, ET, asrc2, adst2, p2, (unsigned*)m2, 1);
  edge_exp<<<blks(ET, TB), TB, 0, stream>>>(
      src, dst, E0, ET, p2, (const unsigned*)m2, denom2, 1);
  edge_scatter<<<blks((long long)ET * OUT_C, TB), TB, 0, stream>>>(
      src, dst, E0, ET, h2, p2, denom2, out2, 1, OUT_C);

  // ================= readout =================
  mean_broadcast<<<OUT_C, 256, 0, stream>>>(out2, b2, out, rows);
}